// RadialBasis_55465207660667
// MI455X (gfx1250) — compile-verified
//
#include <hip/hip_runtime.h>
#include <hip/hip_bf16.h>
#include <stdint.h>

#define LMAX 5
#define NMAX 16
#define NLN 96            // (LMAX+1)*NMAX
#define KSTART 15         // LMAX + PAD(10): single fixed-depth Miller recurrence
#define QUADS_PER_R 24    // NLN / 4 outputs per thread

#if defined(__has_builtin)
#if __has_builtin(__builtin_amdgcn_global_load_async_to_lds_b32)
#define ASYNC_LDS 1
#endif
#endif
#ifndef ASYNC_LDS
#define ASYNC_LDS 0
#endif

typedef __attribute__((address_space(1))) int gas_int;   // global AS pointer elem
typedef __attribute__((address_space(3))) int las_int;   // LDS AS pointer elem

__global__ __launch_bounds__(256) void RadialBasis_kernel(
    const float* __restrict__ r,
    const float* __restrict__ zeros,   // [NMAX, LMAX+1] row-major: idx = n*6 + l
    const float* __restrict__ norms,   // same layout
    float* __restrict__ out,           // [N, LMAX+1, NMAX] flat
    unsigned int nquads)
{
    __shared__ float s_zeros[NLN];
    __shared__ float s_norms[NLN];

    const unsigned int tid = threadIdx.x;
    const unsigned int i4  = blockIdx.x * 256u + tid;

    // ---- stage the tiny tables into LDS via gfx1250 async global->LDS DMA ----
#if ASYNC_LDS
    if (tid < NLN) {
        float* gz = const_cast<float*>(zeros) + tid;
        float* gn = const_cast<float*>(norms) + tid;
        __builtin_amdgcn_global_load_async_to_lds_b32(
            (gas_int*)gz, (las_int*)(s_zeros + tid), 0, 0);
        __builtin_amdgcn_global_load_async_to_lds_b32(
            (gas_int*)gn, (las_int*)(s_norms + tid), 0, 0);
    }
#else
    if (tid < NLN) {
        s_zeros[tid] = zeros[tid];
        s_norms[tid] = norms[tid];
    }
#endif

    // decode which 4 outputs this thread owns: same r, same l, n0..n0+3
    const unsigned int ri = i4 / QUADS_PER_R;
    const unsigned int q  = i4 - ri * QUADS_PER_R;
    const unsigned int l  = q >> 2;          // 0..5
    const unsigned int n0 = (q & 3u) << 2;   // 0,4,8,12

    // overlap: prefetch r while the table DMA is in flight (global_prefetch_b8)
    if (i4 < nquads) __builtin_prefetch(r + ri, 0, 0);

#if ASYNC_LDS
#if defined(__has_builtin) && __has_builtin(__builtin_amdgcn_s_wait_asynccnt)
    __builtin_amdgcn_s_wait_asynccnt(0);
#else
    asm volatile("s_wait_asynccnt 0" ::: "memory");
#endif
#endif
    __syncthreads();

    if (i4 >= nquads) return;

    const float rv = r[ri];

    float u[4], sn[4], fk[4], fkp1[4], ft[4];
#pragma unroll
    for (int j = 0; j < 4; ++j) {
        const unsigned int idx = (n0 + (unsigned)j) * 6u + l;   // [n][l] table layout
        float x = s_zeros[idx] * 0.2f * rv;                     // zeros*r/A, A=5
        x = fmaxf(x, 1e-12f);
        u[j]    = __fdividef(1.0f, x);
        sn[j]   = __sinf(x);
        fk[j]   = 1e-20f;   // Miller seed
        fkp1[j] = 0.0f;
        ft[j]   = 1e-20f;
    }

    // downward Miller recurrence, fixed depth, branch-free capture of f_l
#pragma unroll
    for (int k = KSTART; k >= 1; --k) {
        const float c = (float)(2 * k + 1);
#pragma unroll
        for (int j = 0; j < 4; ++j) {
            const float fkm1 = fmaf(c, u[j] * fk[j], -fkp1[j]);
            if ((k - 1) <= LMAX) {                       // compile-time pruned
                ft[j] = (l == (unsigned)(k - 1)) ? fkm1 : ft[j];
            }
            fkp1[j] = fk[j];
            fk[j]   = fkm1;                              // ends holding f_0
        }
    }

    float ov[4];
#pragma unroll
    for (int j = 0; j < 4; ++j) {
        const unsigned int idx = (n0 + (unsigned)j) * 6u + l;
        const float j0    = sn[j] * u[j];                // sin(x)/x
        const float scale = __fdividef(j0, fk[j]);       // j0 / f0 normalization
        ov[j] = s_norms[idx] * ft[j] * scale;
    }

    // coalesced 128-bit store; float4 slot == out[ri, l, n0..n0+3]
    float4 o; o.x = ov[0]; o.y = ov[1]; o.z = ov[2]; o.w = ov[3];
    *(float4*)(out + (size_t)i4 * 4u) = o;
}

extern "C" void kernel_launch(void* const* d_in, const int* in_sizes, int n_in,
                              void* d_out, int out_size, void* d_ws, size_t ws_size,
                              hipStream_t stream) {
    const float* r     = (const float*)d_in[0];
    const float* zeros = (const float*)d_in[1];
    const float* norms = (const float*)d_in[2];
    float* out = (float*)d_out;

    const unsigned int N      = (unsigned int)in_sizes[0];
    const unsigned int nquads = N * QUADS_PER_R;
    const unsigned int blocks = (nquads + 255u) / 256u;

    RadialBasis_kernel<<<blocks, 256, 0, stream>>>(r, zeros, norms, out, nquads);
}